// Aggregate_68848325754999
// MI455X (gfx1250) — compile-verified
//
#include <hip/hip_runtime.h>

typedef float v2f __attribute__((ext_vector_type(2)));
typedef float v8f __attribute__((ext_vector_type(8)));

#define N_NODES 8192
#define CIN     512
#define COUT    512
#define MAXK    128   // LDS capacity for neighbor list (neibor_num=64 in practice)

// ---------------------------------------------------------------------------
// Kernel 1: per row of A, find the first <=K set entries (ascending column
// order), then mean the corresponding rows of X. One 256-thread block per row;
// wave 0 does the ballot-scan, all threads do the gather-mean (X is L2-resident).
// ---------------------------------------------------------------------------
__global__ void nbr_mean_kernel(const float* __restrict__ X,
                                const int*   __restrict__ A,
                                const int*   __restrict__ pK,
                                float*       __restrict__ nb_mean,
                                int*         __restrict__ cnt_out) {
    __shared__ int s_nbr[MAXK];
    __shared__ int s_cnt;

    const int row = blockIdx.x;
    const int tid = threadIdx.x;
    const int K   = min(*pK, MAXK);

    if (tid < 32) {
        const int  lane = tid;
        const int* arow = A + (size_t)row * N_NODES;
        int found = 0;
        for (int c0 = 0; c0 < N_NODES && found < K; c0 += 32) {
            const int a = arow[c0 + lane];                       // coalesced 128B
            const unsigned m   = __builtin_amdgcn_ballot_w32(a > 0);
            const int      pre = __popc(m & ((1u << lane) - 1u));
            const int      pos = found + pre;
            if ((a > 0) && pos < K) s_nbr[pos] = c0 + lane;      // first-K truncation
            found = min(K, found + __popc(m));
        }
        if (lane == 0) s_cnt = found;
    }
    __syncthreads();

    const int cnt = s_cnt;
    float acc0 = 0.0f, acc1 = 0.0f;                              // cols tid, tid+256
    for (int k = 0; k < cnt; ++k) {
        const float* xr = X + (size_t)s_nbr[k] * CIN;            // L2 hit
        acc0 += xr[tid];
        acc1 += xr[tid + 256];
    }
    const float inv  = 1.0f / (float)max(cnt, 1);
    float*      orow = nb_mean + (size_t)row * CIN;
    orow[tid]       = acc0 * inv;
    orow[tid + 256] = acc1 * inv;
    if (tid == 0) cnt_out[row] = cnt;
}

// ---------------------------------------------------------------------------
// Kernel 2: fused dual GEMM with V_WMMA_F32_16X16X4_F32 (full fp32 precision).
//   out[i,o] = leaky(X[i,:]*W_line[o,:] + b_line[o])
//            + (cnt[i]>0) * leaky(nb_mean[i,:]*W_nb[o,:] + b_nb[o])
// One wave per 16x16 output tile; K-loop over 512 in steps of 4.
// A frag (16x4 MxK): lane l<16 -> A[l][0..1], lane l+16 -> A[l][2..3]
// B frag (4x16 KxN): lane n<16 -> B[0..1][n], lane n+16 -> B[2..3][n]
// Both map to one float2 load: W is [Cout,Cin] so B[k][n] = W[o0+n][k0+k].
// ---------------------------------------------------------------------------
__device__ __forceinline__ float leaky(float x) { return x > 0.0f ? x : 0.01f * x; }

__global__ void fused_gemm_kernel(const float* __restrict__ X,
                                  const float* __restrict__ NBM,
                                  const float* __restrict__ Wl,
                                  const float* __restrict__ bl,
                                  const float* __restrict__ Wn,
                                  const float* __restrict__ bn,
                                  const int*   __restrict__ cnt,
                                  float*       __restrict__ out) {
    const int lane  = threadIdx.x & 31;
    const int wave  = threadIdx.x >> 5;
    const int row0  = (blockIdx.y * 8 + wave) * 16;   // M tile
    const int o0    = blockIdx.x * 16;                // N tile

    const int m     = lane & 15;
    const int khalf = (lane >> 4) * 2;                // 0 or 2

    const float* xrow  = X   + (size_t)(row0 + m) * CIN + khalf;
    const float* nrow  = NBM + (size_t)(row0 + m) * CIN + khalf;
    const float* wlrow = Wl  + (size_t)(o0  + m) * CIN + khalf;
    const float* wnrow = Wn  + (size_t)(o0  + m) * CIN + khalf;

    v8f accL = {};
    v8f accN = {};
#pragma unroll 4
    for (int k0 = 0; k0 < CIN; k0 += 4) {
        const v2f a  = *(const v2f*)(xrow  + k0);
        const v2f bL = *(const v2f*)(wlrow + k0);
        const v2f an = *(const v2f*)(nrow  + k0);
        const v2f bN = *(const v2f*)(wnrow + k0);
        accL = __builtin_amdgcn_wmma_f32_16x16x4_f32(false, a,  false, bL,
                                                     (short)0, accL, false, false);
        accN = __builtin_amdgcn_wmma_f32_16x16x4_f32(false, an, false, bN,
                                                     (short)0, accN, false, false);
    }

    // C/D layout: lane<16 VGPR r -> C[M=r][N=lane]; lane>=16 -> C[M=8+r][N=lane-16]
    const int   col   = o0 + (lane & 15);
    const float biasL = bl[col];
    const float biasN = bn[col];
    const int   rbase = row0 + (lane >> 4) * 8;
#pragma unroll
    for (int r = 0; r < 8; ++r) {
        const int   row = rbase + r;
        const float xi  = leaky(accL[r] + biasL);
        const float xj  = leaky(accN[r] + biasN);
        out[(size_t)row * COUT + col] = xi + ((cnt[row] > 0) ? xj : 0.0f);
    }
}

// ---------------------------------------------------------------------------
extern "C" void kernel_launch(void* const* d_in, const int* in_sizes, int n_in,
                              void* d_out, int out_size, void* d_ws, size_t ws_size,
                              hipStream_t stream) {
    const float* X      = (const float*)d_in[0];
    const int*   A      = (const int*)  d_in[1];
    const float* W_nb   = (const float*)d_in[2];
    const float* b_nb   = (const float*)d_in[3];
    const float* W_line = (const float*)d_in[4];
    const float* b_line = (const float*)d_in[5];
    const int*   nbnum  = (const int*)  d_in[6];   // device scalar (=64)

    float* out    = (float*)d_out;
    float* nbmean = (float*)d_ws;                                  // 8192*512 f32
    int*   cntbuf = (int*)((char*)d_ws + (size_t)N_NODES * CIN * sizeof(float));

    nbr_mean_kernel<<<dim3(N_NODES), dim3(256), 0, stream>>>(
        X, A, nbnum, nbmean, cntbuf);

    fused_gemm_kernel<<<dim3(COUT / 16, N_NODES / (16 * 8)), dim3(256), 0, stream>>>(
        X, nbmean, W_line, b_line, W_nb, b_nb, cntbuf, out);
}